// GraphAttentionLayer_53781580481001
// MI455X (gfx1250) — compile-verified
//
#include <hip/hip_runtime.h>

typedef float v2f __attribute__((ext_vector_type(2)));
typedef float v8f __attribute__((ext_vector_type(8)));
typedef unsigned int u32x4 __attribute__((ext_vector_type(4)));
typedef int i32x4 __attribute__((ext_vector_type(4)));
typedef int i32x8 __attribute__((ext_vector_type(8)));

#define HEADS 8
#define BATCH 8
#define NN    1024
#define KIN   768
#define KOUT  96
// TDM pads 1 DWORD after every 256 DWORDs stored -> effective LDS strides:
#define FSTR  771    // 768 + 3 pads :  3*r distinct banks (mod 64) for r=0..15
#define ADJS  1028   // 1024 + 4 pads:  4*r distinct banks (mod 64) for r=0..15

static __device__ __forceinline__ v8f wmma_f32(v2f a, v2f b, v8f c) {
  // D = A(16x4) * B(4x16) + C, all f32, wave32
  return __builtin_amdgcn_wmma_f32_16x16x4_f32(false, a, false, b, (short)0, c,
                                               false, false);
}

// ---------------------------------------------------------------------------
// Tensor Data Mover: 2D tile (tile_w x tile_h f32) global -> LDS, with
// 1-DWORD pad per 256 DWORDs (pad_interval code 7, pad_amount code 0).
// Element (r,c) lands at LDS dword  r*(tile_w + tile_w/256) + c + (c>>8).
// Issue from ONE wave; tracked on TENSORcnt.
// ---------------------------------------------------------------------------
#if defined(__gfx1250__)
static __device__ __forceinline__ void tdm_load_tile_f32(
    unsigned lds_off_bytes, const void* gptr, unsigned tile_w, unsigned tile_h,
    unsigned row_stride_elems) {
  const unsigned long long ga = (unsigned long long)(size_t)gptr;
  u32x4 g0;
  g0[0] = 1u;                                   // count=1, user descriptor
  g0[1] = lds_off_bytes;                        // lds_addr
  g0[2] = (unsigned)(ga & 0xFFFFFFFFu);         // global_addr[31:0]
  g0[3] = (unsigned)((ga >> 32) & 0x01FFFFFFu)  // global_addr[56:32]
          | (2u << 30);                         // type = 2 ("image")
  i32x8 g1;
  g1[0] = (int)((2u << 16)      // data_size = 4B
                | (1u << 20)    // pad_enable
                | (7u << 22));  // pad_interval = 256 DWORDs (pad_amount=1 DW)
  g1[1] = (int)(tile_w << 16);  // tensor_dim0[15:0] in bits[63:48]
  g1[2] = (int)(tile_h << 16);  // tensor_dim0 hi=0 ; tensor_dim1[15:0]
  g1[3] = (int)(tile_w << 16);  // tensor_dim1 hi=0 ; tile_dim0
  g1[4] = (int)tile_h;          // tile_dim1 ; tile_dim2 = 0
  g1[5] = (int)row_stride_elems;  // tensor_dim0_stride[31:0]
  g1[6] = 0;
  g1[7] = 0;
  const i32x4 z4 = {0, 0, 0, 0};
#if __clang_major__ >= 23
  const i32x8 z8 = {0, 0, 0, 0, 0, 0, 0, 0};
  __builtin_amdgcn_tensor_load_to_lds(g0, g1, z4, z4, z8, 0);
#else
  __builtin_amdgcn_tensor_load_to_lds(g0, g1, z4, z4, 0);
#endif
}
#endif

// Stage a [tile_h x tile_w] f32 tile into padded LDS. Wave 0 uses the TDM;
// fallback (host pass / other targets) is a cooperative scalar copy.
static __device__ __forceinline__ void stage_tile(float* lds, const float* g,
                                                  int tile_w, int tile_h,
                                                  int row_stride, int tid) {
#if defined(__gfx1250__)
  if (tid < 32) {
    tdm_load_tile_f32((unsigned)(size_t)lds, g, (unsigned)tile_w,
                      (unsigned)tile_h, (unsigned)row_stride);
    __builtin_amdgcn_s_wait_tensorcnt(0);
  }
#else
  const int pitch = tile_w + (tile_w >> 8);
  for (int i = tid; i < tile_w * tile_h; i += 256) {
    const int r = i / tile_w, c = i % tile_w;
    lds[r * pitch + c + (c >> 8)] = g[(size_t)r * row_stride + c];
  }
#endif
}

// ---------------------------------------------------------------------------
// Kernel 1: h[b,h,n,o] = feat[b,n,:] @ W[h,:,:]; t = tanh(h);
//           a_src[b,h,n] = t . w_src[h]; a_dst[b,h,n] = t . w_dst[h]
// grid = B * N/16 ; block = 256 (8 waves, wave == head)
// ---------------------------------------------------------------------------
__global__ __launch_bounds__(256) void gat_proj(
    const float* __restrict__ feat, const float* __restrict__ W,
    const float* __restrict__ wsrc, const float* __restrict__ wdst,
    float* __restrict__ h_ws, float* __restrict__ asrc_ws,
    float* __restrict__ adst_ws) {
  __shared__ float lds_feat[16 * FSTR + 4];
  __shared__ float lds_as[HEADS * 16];
  __shared__ float lds_ad[HEADS * 16];

  const int b  = blockIdx.x >> 6;
  const int n0 = (blockIdx.x & 63) << 4;
  const int tid = threadIdx.x;

  stage_tile(lds_feat, feat + ((size_t)b * NN + n0) * KIN, KIN, 16, KIN, tid);
  if (tid < HEADS * 16) { lds_as[tid] = 0.f; lds_ad[tid] = 0.f; }
  __syncthreads();

  const int head = tid >> 5, lane = tid & 31;
  const int nl = lane & 15, half = lane >> 4;
  const float* Wh = W + (size_t)head * KIN * KOUT;

  v8f acc[6] = {};
  for (int k0 = 0; k0 < KIN; k0 += 4) {
    const int kk = k0 + 2 * half;
    const int ai = nl * FSTR + kk + (kk >> 8);
    v2f a;
    a.x = lds_feat[ai];
    a.y = lds_feat[ai + 1];
    const float* wp = Wh + (size_t)kk * KOUT + nl;
#pragma unroll
    for (int j = 0; j < 6; ++j) {
      v2f bm;
      bm.x = wp[j * 16];
      bm.y = wp[j * 16 + KOUT];
      acc[j] = wmma_f32(a, bm, acc[j]);
    }
  }

  float wsc[6], wdc[6];
#pragma unroll
  for (int j = 0; j < 6; ++j) {
    wsc[j] = wsrc[head * KOUT + j * 16 + nl];
    wdc[j] = wdst[head * KOUT + j * 16 + nl];
  }
  float ps[8] = {0.f}, pd[8] = {0.f};
  float* hp = h_ws + (((size_t)b * HEADS + head) * NN + n0) * KOUT;
#pragma unroll
  for (int j = 0; j < 6; ++j)
#pragma unroll
    for (int r = 0; r < 8; ++r) {
      const float hv = acc[j][r];
      hp[(size_t)(half * 8 + r) * KOUT + j * 16 + nl] = hv;
      const float t = tanhf(hv);
      ps[r] += t * wsc[j];
      pd[r] += t * wdc[j];
    }
#pragma unroll
  for (int r = 0; r < 8; ++r) {
    atomicAdd(&lds_as[head * 16 + half * 8 + r], ps[r]);
    atomicAdd(&lds_ad[head * 16 + half * 8 + r], pd[r]);
  }
  __syncthreads();
  if (tid < HEADS * 16) {
    const int hh = tid >> 4, rr = tid & 15;
    asrc_ws[((size_t)b * HEADS + hh) * NN + n0 + rr] = lds_as[tid];
    adst_ws[((size_t)b * HEADS + hh) * NN + n0 + rr] = lds_ad[tid];
  }
}

// ---------------------------------------------------------------------------
// Kernel 2: gate[row, j] = sigmoid(feat[row,:] . Hw[j,:] + Hb[j])
// grid = (B*N)/16 ; block = 256 (8 waves, wave == 96-col strip)
// ---------------------------------------------------------------------------
__global__ __launch_bounds__(256) void gat_gate(
    const float* __restrict__ feat, const float* __restrict__ Hw,
    const float* __restrict__ Hb, float* __restrict__ gate_ws) {
  __shared__ float lds_feat[16 * FSTR + 4];
  const size_t g0 = (size_t)blockIdx.x * 16;
  const int tid = threadIdx.x;

  stage_tile(lds_feat, feat + g0 * KIN, KIN, 16, KIN, tid);
  __syncthreads();

  const int wv = tid >> 5, lane = tid & 31;
  const int nl = lane & 15, half = lane >> 4;
  const int c0 = wv * 96;

  v8f acc[6] = {};
  for (int k0 = 0; k0 < KIN; k0 += 4) {
    const int kk = k0 + 2 * half;
    const int ai = nl * FSTR + kk + (kk >> 8);
    v2f a;
    a.x = lds_feat[ai];
    a.y = lds_feat[ai + 1];
#pragma unroll
    for (int j = 0; j < 6; ++j) {
      const int col = c0 + j * 16 + nl;
      const float* hwp = Hw + (size_t)col * KIN + kk;  // Hw transposed access
      v2f bm;
      bm.x = hwp[0];
      bm.y = hwp[1];
      acc[j] = wmma_f32(a, bm, acc[j]);
    }
  }
#pragma unroll
  for (int j = 0; j < 6; ++j)
#pragma unroll
    for (int r = 0; r < 8; ++r) {
      const int col = c0 + j * 16 + nl;
      const float v = acc[j][r] + Hb[col];
      gate_ws[(g0 + half * 8 + r) * KIN + col] = 1.f / (1.f + __expf(-v));
    }
}

// ---------------------------------------------------------------------------
// Kernel 3: fused mask -> leakyrelu -> softmax -> P@h -> bias/ELU/gate blend
// grid = B * N/16 ; block = 256 (8 waves = 8 heads share one adj tile)
// ---------------------------------------------------------------------------
__global__ __launch_bounds__(256) void gat_attn(
    const float* __restrict__ feat, const float* __restrict__ adj,
    const float* __restrict__ bias, const float* __restrict__ h_ws,
    const float* __restrict__ asrc_ws, const float* __restrict__ adst_ws,
    const float* __restrict__ gate_ws, float* __restrict__ out) {
  __shared__ float lds_adj[16 * ADJS];
  __shared__ float lds_rmax[HEADS * 16];
  __shared__ float lds_rsum[HEADS * 16];

  const int b  = blockIdx.x >> 6;
  const int n0 = (blockIdx.x & 63) << 4;
  const int tid = threadIdx.x;

  stage_tile(lds_adj, adj + ((size_t)b * NN + n0) * NN, NN, 16, NN, tid);
  __syncthreads();

  const int head = tid >> 5, lane = tid & 31;
  const int nl = lane & 15, half = lane >> 4;
  const size_t base = ((size_t)b * HEADS + head) * NN;
  const float* adst = adst_ws + base;
  const float* hp = h_ws + base * KOUT;
  __builtin_prefetch(hp, 0, 1);  // global_prefetch: pull L2-hot h panel

  // ---- pass 1: per-row max of masked leaky scores ----
  float asr[16];
#pragma unroll
  for (int r = 0; r < 16; ++r) asr[r] = asrc_ws[base + n0 + r];
  float mx[16];
#pragma unroll
  for (int r = 0; r < 16; ++r) mx[r] = -3.0e38f;
  for (int m = lane; m < NN; m += 32) {
    const float ad = adst[m];
    const int mi = m + (m >> 8);
#pragma unroll
    for (int r = 0; r < 16; ++r) {
      float s = asr[r] + ad;
      s = s >= 0.f ? s : 0.2f * s;
      s = (lds_adj[r * ADJS + mi] != 0.f) ? s : -999.f;
      mx[r] = fmaxf(mx[r], s);
    }
  }
#pragma unroll
  for (int off = 16; off > 0; off >>= 1)
#pragma unroll
    for (int r = 0; r < 16; ++r)
      mx[r] = fmaxf(mx[r], __shfl_xor(mx[r], off, 32));
  if (lane == 0) {
#pragma unroll
    for (int r = 0; r < 16; ++r) lds_rmax[head * 16 + r] = mx[r];
  }
  __syncthreads();

  // ---- pass 2: P fragment built in-register, WMMA against h ----
  const float my_as = asrc_ws[base + n0 + nl];
  const float my_mx = lds_rmax[head * 16 + nl];

  v8f acc[6] = {};
  float rs = 0.f;
  for (int m0 = 0; m0 < NN; m0 += 4) {
    const int mm = m0 + 2 * half;
    const int mi = nl * ADJS + mm + (mm >> 8);  // mm even -> mm+1 same block
    float s0 = my_as + adst[mm];
    float s1 = my_as + adst[mm + 1];
    s0 = s0 >= 0.f ? s0 : 0.2f * s0;
    s1 = s1 >= 0.f ? s1 : 0.2f * s1;
    s0 = (lds_adj[mi] != 0.f) ? s0 : -999.f;
    s1 = (lds_adj[mi + 1] != 0.f) ? s1 : -999.f;
    const float p0 = __expf(s0 - my_mx);
    const float p1 = __expf(s1 - my_mx);
    rs += p0 + p1;
    v2f a;
    a.x = p0;
    a.y = p1;
    const float* hk = hp + (size_t)mm * KOUT + nl;
#pragma unroll
    for (int j = 0; j < 6; ++j) {
      v2f bm;
      bm.x = hk[j * 16];
      bm.y = hk[j * 16 + KOUT];
      acc[j] = wmma_f32(a, bm, acc[j]);
    }
  }
  rs += __shfl_xor(rs, 16, 32);
  if (half == 0) lds_rsum[head * 16 + nl] = rs;
  __syncthreads();

  // ---- epilogue: normalize, bias, ELU, gated residual blend ----
#pragma unroll
  for (int j = 0; j < 6; ++j)
#pragma unroll
    for (int r = 0; r < 8; ++r) {
      const int row = half * 8 + r;
      const int col = j * 16 + nl;
      float v = acc[j][r] / lds_rsum[head * 16 + row] + bias[col];
      const float e = v > 0.f ? v : (__expf(v) - 1.f);
      const size_t gi = ((size_t)b * NN + n0 + row) * KIN + head * KOUT + col;
      const float g = gate_ws[gi];
      out[gi] = g * e + (1.f - g) * feat[gi];
    }
}

// ---------------------------------------------------------------------------
extern "C" void kernel_launch(void* const* d_in, const int* in_sizes, int n_in,
                              void* d_out, int out_size, void* d_ws,
                              size_t ws_size, hipStream_t stream) {
  const float* feat = (const float*)d_in[0];
  const float* adj  = (const float*)d_in[1];
  const float* W    = (const float*)d_in[2];
  const float* bias = (const float*)d_in[3];
  const float* wsrc = (const float*)d_in[4];
  const float* wdst = (const float*)d_in[5];
  const float* Hw   = (const float*)d_in[6];
  const float* Hb   = (const float*)d_in[7];
  float* out = (float*)d_out;

  // workspace partition (~51 MB)
  float* h_ws  = (float*)d_ws;                                  // B*H*N*96
  float* asrc  = h_ws + (size_t)BATCH * HEADS * NN * KOUT;      // B*H*N
  float* adstp = asrc + (size_t)BATCH * HEADS * NN;             // B*H*N
  float* gate  = adstp + (size_t)BATCH * HEADS * NN;            // B*N*768

  const dim3 blk(256);
  gat_proj<<<dim3(BATCH * (NN / 16)), blk, 0, stream>>>(feat, W, wsrc, wdst,
                                                        h_ws, asrc, adstp);
  gat_gate<<<dim3((BATCH * NN) / 16), blk, 0, stream>>>(feat, Hw, Hb, gate);
  gat_attn<<<dim3(BATCH * (NN / 16)), blk, 0, stream>>>(
      feat, adj, bias, h_ws, asrc, adstp, gate, out);
}